// TimeVaryingLinear_59622736003997
// MI455X (gfx1250) — compile-verified
//
#include <hip/hip_runtime.h>
#include <math.h>

// ---------------------------------------------------------------------------
// TimeVaryingLinear, fused:
//   v[b,d] = sum_o ( sum_x cw[b,x] * softplus(table[c0,c1,o]) ) * W[o,d]
//   out[b,i] = sum_d x[b,i,d] * v[b,d]
// Batched GEMV, HBM-bound (536 MB of x @ 23.3 TB/s ~ 23 us floor).
// WMMA trick: A[m,k] = x[b,i0+m,k] * v[b,k], B = all-ones -> D row-sum = dot.
// ---------------------------------------------------------------------------

typedef __attribute__((ext_vector_type(2))) float v2f;
typedef __attribute__((ext_vector_type(8))) float v8f;

#define I_  256
#define D_  64
#define X_  4
#define O_  4
#define T0_ 3
#define T1_ 15

__global__ __launch_bounds__(512) void tvl_wmma_kernel(
    const float* __restrict__ x,      // (B, I, D)
    const int*   __restrict__ coords, // (B, X, 2)
    const float* __restrict__ cw,     // (B, X)
    const float* __restrict__ table,  // (T0, T1, O)
    const float* __restrict__ W,      // (O, D)
    float*       __restrict__ out)    // (B, I)
{
    __shared__ float vsh[D_];

    const int b   = blockIdx.x;
    const int tid = threadIdx.x;

    // ---- Stage 1: per-batch vector v[b,:] (threads 0..63, one per d) ----
    if (tid < D_) {
        float slw[O_] = {0.f, 0.f, 0.f, 0.f};
        #pragma unroll
        for (int p = 0; p < X_; ++p) {
            const int   c0 = coords[(b * X_ + p) * 2 + 0];
            const int   c1 = coords[(b * X_ + p) * 2 + 1];
            const float w  = cw[b * X_ + p];
            const float* t = table + (c0 * T1_ + c1) * O_;
            #pragma unroll
            for (int o = 0; o < O_; ++o) {
                const float tv = t[o];
                // stable softplus via hardware transcendentals:
                //   softplus(x) = max(x,0) + log(1 + exp(-|x|))
                // log argument is in (1,2] -> perfectly conditioned, v_exp_f32 /
                // v_log_f32 (~1 ulp) keep absolute error ~1e-7 on values ~0.7.
                const float e  = __expf(-fabsf(tv));
                const float sp = fmaxf(tv, 0.f) + __logf(1.0f + e);
                slw[o] += w * sp;
            }
        }
        float v = 0.f;
        #pragma unroll
        for (int o = 0; o < O_; ++o) v += slw[o] * W[o * D_ + tid];
        vsh[tid] = v;
    }
    __syncthreads();

    // ---- Stage 2: WMMA row-sum dot products ----
    // wave w handles rows i0 = 16*w .. 16*w+15 of x[b]
    const int lane = tid & 31;
    const int wave = tid >> 5;
    const int i0   = wave * 16;
    const int m    = lane & 15;   // A-matrix row M (documented 16x4 f32 layout)
    const int h    = lane >> 4;   // lane-half: K offset 0 (lo) / 2 (hi)

    const float* __restrict__ row = x + ((size_t)b * I_ + (i0 + m)) * D_;

    // A fragments for 16 K-chunks of 4: lane supplies K = 4c+2h, 4c+2h+1
    v2f a[16];
    #pragma unroll
    for (int c = 0; c < 16; ++c) {
        const int k0 = 4 * c + 2 * h;
        const float2 xv = *(const float2*)(row + k0);   // global_load_b64
        v2f av;
        av[0] = xv.x * vsh[k0];
        av[1] = xv.y * vsh[k0 + 1];
        a[c] = av;
    }

    v2f ones;
    ones[0] = 1.0f;
    ones[1] = 1.0f;

    v8f acc = {};
    #pragma unroll
    for (int c = 0; c < 16; ++c) {
        // D = A x B + C ; B == 1 everywhere -> D[m,n] = sum_k A[m,k] (all n equal)
        acc = __builtin_amdgcn_wmma_f32_16x16x4_f32(
            /*neg_a=*/false, a[c],
            /*neg_b=*/false, ones,
            /*c_mod=*/(short)0, acc,
            /*reuse_a=*/false, /*reuse_b=*/false);
    }

    // ---- Store: C/D layout: lane N=0 (lanes 0,16) holds rows h*8 + r in acc[r]
    if (m == 0) {
        float* dst = out + (size_t)b * I_ + i0 + h * 8;
        float4 lo = make_float4(acc[0], acc[1], acc[2], acc[3]);
        float4 hi = make_float4(acc[4], acc[5], acc[6], acc[7]);
        *(float4*)(dst + 0) = lo;   // global_store_b128
        *(float4*)(dst + 4) = hi;   // global_store_b128
    }
}

extern "C" void kernel_launch(void* const* d_in, const int* in_sizes, int n_in,
                              void* d_out, int out_size, void* d_ws, size_t ws_size,
                              hipStream_t stream) {
    const float* x      = (const float*)d_in[0];
    const int*   coords = (const int*)d_in[1];
    const float* cw     = (const float*)d_in[2];
    const float* table  = (const float*)d_in[3];
    const float* W      = (const float*)d_in[4];
    float*       out    = (float*)d_out;

    const int B = in_sizes[0] / (I_ * D_);   // 8192 for the reference shapes

    tvl_wmma_kernel<<<B, 512, 0, stream>>>(x, coords, cw, table, W, out);
}